// LocalInteraction_37417755082831
// MI455X (gfx1250) — compile-verified
//
#include <hip/hip_runtime.h>

// ---------------------------------------------------------------------------
// MI455X (gfx1250) implementation of the LocalInteraction layer.
// f32 end-to-end (matches reference numerics); all GEMMs on V_WMMA_F32_16X16X4_F32.
// Bound by the 184M-element segment-sum scatter (~740MB => ~32us at 23.3TB/s),
// so GEMM precision/throughput is not the limiter and f32 WMMA is the right op.
// ---------------------------------------------------------------------------

typedef float v2f __attribute__((ext_vector_type(2)));
typedef float v8f __attribute__((ext_vector_type(8)));

// f32 WMMA: A 16x4, B 4x16, C/D 16x16.  8-arg VOP3P form:
// (neg_a, A, neg_b, B, c_mod, C, reuse_a, reuse_b)
__device__ __forceinline__ v8f wmma4(v2f a, v2f b, v8f c) {
  return __builtin_amdgcn_wmma_f32_16x16x4_f32(false, a, false, b, (short)0, c,
                                               false, false);
}

__device__ __forceinline__ float swishf(float x, float a, float b) {
  return a * x * (1.0f / (1.0f + __expf(-b * x)));
}

struct ResP {
  const float *W1, *W2, *Wd;            // [128][128] row-major (out_feat, in_feat)
  const float *a1, *a2, *a3;            // scalars (1-elem arrays)
  const float *b1, *b2, *b3;
  const float *c1, *c2, *cd;            // [128]
};

#define LDA 132  // padded LDS row stride (floats) to dodge bank conflicts

// One wave computes a 16(M rows from LDS) x 16(N feats) tile of  X @ W^T,
// K=128 via 32 chained f32 WMMAs.  B[k][n] = W[n][k]  (Wrow = W + n*128).
// A frag: lane<16 -> (M=lane, K=k0..k0+1); lane>=16 -> (M=lane-16, K=k0+2..k0+3).
__device__ __forceinline__ v8f wave_gemm128(const float* As, const float* Wrow,
                                            int mrow, int kh) {
  v8f acc = {};
#pragma unroll
  for (int k0 = 0; k0 < 128; k0 += 4) {
    v2f a; a.x = As[mrow * LDA + k0 + 2 * kh]; a.y = As[mrow * LDA + k0 + 2 * kh + 1];
    v2f b; b.x = Wrow[k0 + 2 * kh];            b.y = Wrow[k0 + 2 * kh + 1];
    acc = wmma4(a, b, acc);
  }
  return acc;
}

// ---------------------------------------------------------------------------
// ResMLP: out = resmlp(x [+ add1 [+ add2]])   (16 nodes per block, 8 waves)
// ---------------------------------------------------------------------------
__global__ __launch_bounds__(256) void resmlp_kernel(
    const float* __restrict__ xin, const float* __restrict__ add1,
    const float* __restrict__ add2, float* __restrict__ out, ResP p) {
  __shared__ float Xs[16 * 128];   // raw input (residual)
  __shared__ float As[16 * LDA];   // activation buffer A
  __shared__ float Bs[16 * LDA];   // activation buffer B

  const int tid  = threadIdx.x;
  const int lane = tid & 31, wave = tid >> 5;
  const int row0 = blockIdx.x * 16;

  const float a1 = p.a1[0], b1 = p.b1[0];
  const float a2 = p.a2[0], b2 = p.b2[0];
  const float a3 = p.a3[0], b3 = p.b3[0];

  for (int i = tid; i < 16 * 128; i += 256) {
    const int rr = i >> 7, cc = i & 127;
    float v = xin[(size_t)(row0 + rr) * 128 + cc];
    if (add1) v += add1[(size_t)(row0 + rr) * 128 + cc];
    if (add2) v += add2[(size_t)(row0 + rr) * 128 + cc];
    Xs[rr * 128 + cc] = v;
    As[rr * LDA + cc] = swishf(v, a1, b1);
  }
  __syncthreads();

  const int mrow = lane & 15;     // N index within tile (also A's M row)
  const int kh   = lane >> 4;     // K-half selector
  const int n    = wave * 16 + mrow;  // global output feature

  // GEMM1: h1 = swish1(x) @ W1^T + c1 ; Bs = swish2(h1)
  v8f acc = wave_gemm128(As, p.W1 + (size_t)n * 128, mrow, kh);
  {
    const float c1 = p.c1[n];
#pragma unroll
    for (int r = 0; r < 8; ++r) {
      const int m = r + 8 * kh;
      Bs[m * LDA + n] = swishf(acc[r] + c1, a2, b2);
    }
  }
  __syncthreads();

  // GEMM2: h2 = Bs @ W2^T + c2 ; h = x + h2 ; As = swish3(h)
  acc = wave_gemm128(Bs, p.W2 + (size_t)n * 128, mrow, kh);
  {
    const float c2 = p.c2[n];
#pragma unroll
    for (int r = 0; r < 8; ++r) {
      const int m = r + 8 * kh;
      const float h = Xs[m * 128 + n] + acc[r] + c2;
      As[m * LDA + n] = swishf(h, a3, b3);
    }
  }
  __syncthreads();

  // GEMM3: out = As @ Wd^T + cd
  acc = wave_gemm128(As, p.Wd + (size_t)n * 128, mrow, kh);
  {
    const float cd = p.cd[n];
#pragma unroll
    for (int r = 0; r < 8; ++r) {
      const int m = r + 8 * kh;
      out[(size_t)(row0 + m) * 128 + n] = acc[r] + cd;
    }
  }
}

// ---------------------------------------------------------------------------
// Edge stage: 16 edges per block.
//   rho_l[e][k] (Bernstein*f_cut), Grho = rho @ G_l^T (WMMA, K=16),
//   per_nbr[e,f,m] = feat_l[src,f] * Grho[e,f] * y_l[e,m]  -> atomic scatter.
// quants layout: q_l[node][m][feat]  (so q0 doubles as s_term [N,128]).
// ---------------------------------------------------------------------------
__device__ const float BINOM16[16] = {1.f, 15.f, 105.f, 455.f, 1365.f, 3003.f,
                                      5005.f, 6435.f, 6435.f, 5005.f, 3003.f,
                                      1365.f, 455.f, 105.f, 15.f, 1.f};

__global__ __launch_bounds__(256) void edge_kernel(
    const float* __restrict__ r_ij, const int* __restrict__ nbrs,
    const float* __restrict__ feat0, const float* __restrict__ feat1,
    const float* __restrict__ feat2, const float* __restrict__ G0,
    const float* __restrict__ G1, const float* __restrict__ G2,
    const float* __restrict__ gi0, const float* __restrict__ gi1,
    const float* __restrict__ gi2, float* __restrict__ q0,
    float* __restrict__ q1, float* __restrict__ q2) {
  __shared__ float rho0[16][16], rho1[16][16], rho2[16][16];
  __shared__ float f0S[16][128], f1S[16][128], f2S[16][128];
  __shared__ float y1S[16][3], y2S[16][5];
  __shared__ float rS[16];
  __shared__ int srcS[16], dstS[16];

  const int tid = threadIdx.x;
  const int e0  = blockIdx.x * 16;

  if (tid < 16) {
    const int e = e0 + tid;
    const float rx = r_ij[(size_t)e * 3 + 0];
    const float ry = r_ij[(size_t)e * 3 + 1];
    const float rz = r_ij[(size_t)e * 3 + 2];
    const float r  = sqrtf(rx * rx + ry * ry + rz * rz + 3e-15f);
    rS[tid] = r;
    const float ir = 1.0f / r;
    const float ux = rx * ir, uy = ry * ir, uz = rz * ir;
    y1S[tid][0] = uy; y1S[tid][1] = uz; y1S[tid][2] = ux;
    const float s3 = 1.7320508075688772f;
    y2S[tid][0] = s3 * ux * uy;
    y2S[tid][1] = s3 * uy * uz;
    y2S[tid][2] = 0.5f * (3.0f * uz * uz - 1.0f);
    y2S[tid][3] = s3 * ux * uz;
    y2S[tid][4] = 0.5f * s3 * (ux * ux - uy * uy);
    dstS[tid] = nbrs[(size_t)e * 2 + 0];
    srcS[tid] = nbrs[(size_t)e * 2 + 1];
  }
  __syncthreads();

  {  // rho: one (edge,k) pair per thread, all three l channels
    const int e = tid >> 4, k = tid & 15;
    const float r  = rS[e];
    float fc = 0.0f;
    if (r < 5.0f) {
      const float denom = (5.0f - r) * (5.0f + r);
      fc = __expf(-(r * r) / denom);
    }
    const float bk = BINOM16[k] * fc;
    const float kf = (float)k, kr = (float)(15 - k);
    float g, ee;
    g = log1pf(__expf(gi0[0])); ee = __expf(-g * r);
    rho0[e][k] = bk * __powf(ee, kf) * __powf(1.0f - ee, kr);
    g = log1pf(__expf(gi1[0])); ee = __expf(-g * r);
    rho1[e][k] = bk * __powf(ee, kf) * __powf(1.0f - ee, kr);
    g = log1pf(__expf(gi2[0])); ee = __expf(-g * r);
    rho2[e][k] = bk * __powf(ee, kf) * __powf(1.0f - ee, kr);
  }

  for (int i = tid; i < 16 * 128; i += 256) {  // gather source-node features
    const int e = i >> 7, f = i & 127;
    const size_t s = (size_t)srcS[e] * 128 + f;
    f0S[e][f] = feat0[s];
    f1S[e][f] = feat1[s];
    f2S[e][f] = feat2[s];
  }
  __syncthreads();

  const int lane = tid & 31, wave = tid >> 5;
  const int mrow = lane & 15, kh = lane >> 4;
  const int n = wave * 16 + mrow;  // feature index for this lane's column

  // ---- l = 0 ----
  {
    const float* Gr = G0 + (size_t)n * 16;
    v8f acc = {};
#pragma unroll
    for (int k0 = 0; k0 < 16; k0 += 4) {
      v2f a; a.x = rho0[mrow][k0 + 2 * kh]; a.y = rho0[mrow][k0 + 2 * kh + 1];
      v2f b; b.x = Gr[k0 + 2 * kh];         b.y = Gr[k0 + 2 * kh + 1];
      acc = wmma4(a, b, acc);
    }
#pragma unroll
    for (int r = 0; r < 8; ++r) {
      const int e = r + 8 * kh;
      const float val = acc[r] * f0S[e][n];
      atomicAdd(q0 + (size_t)dstS[e] * 128 + n, val);
    }
  }
  // ---- l = 1 ----
  {
    const float* Gr = G1 + (size_t)n * 16;
    v8f acc = {};
#pragma unroll
    for (int k0 = 0; k0 < 16; k0 += 4) {
      v2f a; a.x = rho1[mrow][k0 + 2 * kh]; a.y = rho1[mrow][k0 + 2 * kh + 1];
      v2f b; b.x = Gr[k0 + 2 * kh];         b.y = Gr[k0 + 2 * kh + 1];
      acc = wmma4(a, b, acc);
    }
#pragma unroll
    for (int r = 0; r < 8; ++r) {
      const int e = r + 8 * kh;
      const float val = acc[r] * f1S[e][n];
      float* base = q1 + (size_t)dstS[e] * 3 * 128 + n;
#pragma unroll
      for (int m = 0; m < 3; ++m) atomicAdd(base + m * 128, val * y1S[e][m]);
    }
  }
  // ---- l = 2 ----
  {
    const float* Gr = G2 + (size_t)n * 16;
    v8f acc = {};
#pragma unroll
    for (int k0 = 0; k0 < 16; k0 += 4) {
      v2f a; a.x = rho2[mrow][k0 + 2 * kh]; a.y = rho2[mrow][k0 + 2 * kh + 1];
      v2f b; b.x = Gr[k0 + 2 * kh];         b.y = Gr[k0 + 2 * kh + 1];
      acc = wmma4(a, b, acc);
    }
#pragma unroll
    for (int r = 0; r < 8; ++r) {
      const int e = r + 8 * kh;
      const float val = acc[r] * f2S[e][n];
      float* base = q2 + (size_t)dstS[e] * 5 * 128 + n;
#pragma unroll
      for (int m = 0; m < 5; ++m) atomicAdd(base + m * 128, val * y2S[e][m]);
    }
  }
}

// ---------------------------------------------------------------------------
// Mix stage: rows = (node,m) pairs of q_l (layout [n][m][128]).
// t1 = Q @ mixA^T, t2 = Q @ mixB^T (shared A fragment, dual accumulators),
// inv[n,i] += sum_m t1*t2  via f32 atomics.
// ---------------------------------------------------------------------------
__global__ __launch_bounds__(256) void mix_kernel(const float* __restrict__ q,
                                                  const float* __restrict__ mixA,
                                                  const float* __restrict__ mixB,
                                                  float* __restrict__ inv, int mL) {
  __shared__ float Qs[16 * LDA];
  const int tid  = threadIdx.x;
  const int row0 = blockIdx.x * 16;

  for (int i = tid; i < 16 * 128; i += 256) {
    const int rr = i >> 7, cc = i & 127;
    Qs[rr * LDA + cc] = q[(size_t)(row0 + rr) * 128 + cc];
  }
  __syncthreads();

  const int lane = tid & 31, wave = tid >> 5;
  const int mrow = lane & 15, kh = lane >> 4;
  const int n = wave * 16 + mrow;
  const float* wa = mixA + (size_t)n * 128;
  const float* wb = mixB + (size_t)n * 128;

  v8f acc1 = {}, acc2 = {};
#pragma unroll
  for (int k0 = 0; k0 < 128; k0 += 4) {
    v2f a;  a.x  = Qs[mrow * LDA + k0 + 2 * kh]; a.y  = Qs[mrow * LDA + k0 + 2 * kh + 1];
    v2f ba; ba.x = wa[k0 + 2 * kh];              ba.y = wa[k0 + 2 * kh + 1];
    v2f bb; bb.x = wb[k0 + 2 * kh];              bb.y = wb[k0 + 2 * kh + 1];
    acc1 = wmma4(a, ba, acc1);
    acc2 = wmma4(a, bb, acc2);
  }
#pragma unroll
  for (int r = 0; r < 8; ++r) {
    const int row  = row0 + r + 8 * kh;
    const int node = row / mL;
    atomicAdd(inv + (size_t)node * 128 + n, acc1[r] * acc2[r]);
  }
}

__global__ void zero_kernel(float* __restrict__ p, size_t count) {
  for (size_t i = (size_t)blockIdx.x * blockDim.x + threadIdx.x; i < count;
       i += (size_t)gridDim.x * blockDim.x)
    p[i] = 0.0f;
}

// ---------------------------------------------------------------------------
// Launch.  Input order assumption: top-level setup_inputs() dict order
// (xyz, x_tilde, nbrs, r_ij, params...), with `params` flattened pytree-style
// (sorted keys): G_0,G_1,G_2, gamma_inv_0..2, mix_1_1, mix_1_2, mix_2_1,
// mix_2_2, resmlp_0, resmlp_1, resmlp_2, resmlp_c, resmlp_l; each resmlp
// sorted: W1,W2,Wd,a1,a2,a3,b1,b2,b3,c1,c2,cd.
// ---------------------------------------------------------------------------
extern "C" void kernel_launch(void* const* d_in, const int* in_sizes, int n_in,
                              void* d_out, int out_size, void* d_ws,
                              size_t ws_size, hipStream_t stream) {
  const float* x_tilde = (const float*)d_in[1];
  const int*   nbrs    = (const int*)d_in[2];   // harness contract: ints -> const int*
  const float* r_ij    = (const float*)d_in[3];

  const float* G0  = (const float*)d_in[4];
  const float* G1  = (const float*)d_in[5];
  const float* G2  = (const float*)d_in[6];
  const float* gi0 = (const float*)d_in[7];
  const float* gi1 = (const float*)d_in[8];
  const float* gi2 = (const float*)d_in[9];
  const float* mix11 = (const float*)d_in[10];
  const float* mix12 = (const float*)d_in[11];
  const float* mix21 = (const float*)d_in[12];
  const float* mix22 = (const float*)d_in[13];

  auto unpack = [&](int base) {
    ResP p;
    p.W1 = (const float*)d_in[base + 0];
    p.W2 = (const float*)d_in[base + 1];
    p.Wd = (const float*)d_in[base + 2];
    p.a1 = (const float*)d_in[base + 3];
    p.a2 = (const float*)d_in[base + 4];
    p.a3 = (const float*)d_in[base + 5];
    p.b1 = (const float*)d_in[base + 6];
    p.b2 = (const float*)d_in[base + 7];
    p.b3 = (const float*)d_in[base + 8];
    p.c1 = (const float*)d_in[base + 9];
    p.c2 = (const float*)d_in[base + 10];
    p.cd = (const float*)d_in[base + 11];
    return p;
  };
  const ResP r0 = unpack(14), r1 = unpack(26), r2 = unpack(38);
  const ResP rc = unpack(50), rl = unpack(62);

  const int N = in_sizes[1] / 128;   // 10000
  const int E = in_sizes[3] / 3;     // 160000
  const size_t NF = (size_t)N * 128;

  // Workspace layout (floats): c | f0 | f1 | f2 | q0 | q1(x3) | q2(x5) | inv
  float* ws     = (float*)d_ws;
  float* c_term = ws;
  float* f0     = ws + 1 * NF;
  float* f1     = ws + 2 * NF;
  float* f2     = ws + 3 * NF;
  float* q0     = ws + 4 * NF;
  float* q1     = ws + 5 * NF;
  float* q2     = ws + 8 * NF;
  float* inv    = ws + 13 * NF;      // total 14*NF floats (~72 MB)

  // 1) clear accumulators (q0,q1,q2,inv = 10*NF floats)
  zero_kernel<<<4096, 256, 0, stream>>>(q0, 10 * NF);

  // 2) node ResMLPs on x_tilde
  const int nodeBlocks = N / 16;
  resmlp_kernel<<<nodeBlocks, 256, 0, stream>>>(x_tilde, nullptr, nullptr, c_term, rc);
  resmlp_kernel<<<nodeBlocks, 256, 0, stream>>>(x_tilde, nullptr, nullptr, f0, r0);
  resmlp_kernel<<<nodeBlocks, 256, 0, stream>>>(x_tilde, nullptr, nullptr, f1, r1);
  resmlp_kernel<<<nodeBlocks, 256, 0, stream>>>(x_tilde, nullptr, nullptr, f2, r2);

  // 3) edge stage: basis GEMMs + scatter into quants
  edge_kernel<<<E / 16, 256, 0, stream>>>(r_ij, nbrs, f0, f1, f2, G0, G1, G2,
                                          gi0, gi1, gi2, q0, q1, q2);

  // 4) mix stages -> inv
  mix_kernel<<<(N * 3) / 16, 256, 0, stream>>>(q1, mix11, mix12, inv, 3);
  mix_kernel<<<(N * 5) / 16, 256, 0, stream>>>(q2, mix21, mix22, inv, 5);

  // 5) final ResMLP on (c_term + s_term(=q0) + inv) -> d_out
  resmlp_kernel<<<nodeBlocks, 256, 0, stream>>>(c_term, q0, inv, (float*)d_out, rl);
}